// CTLSTM_60017872995025
// MI455X (gfx1250) — compile-verified
//
#include <hip/hip_runtime.h>
#include <math.h>

// CTLSTM forward for MI455X (gfx1250, wave32).
// B=64, T=2048, H=64, F=32. d_in = 96, 7H = 448.
// 4 workgroups x 16 batch rows. W_rec / W_int pre-swizzled into WMMA
// B-fragment layout in LDS (~218 KB of the WGP's 320 KB). x_t staging is
// software-pipelined (load x_{t+1} during step t, prefetch x_{t+2}) so no
// global-load wait sits on the recurrence's critical path.

#define HIDDEN 64
#define NF     32
#define BATCH  64
#define SEQ    2048
#define DIN    96      // NF + HIDDEN
#define NOUT   448     // 7 * HIDDEN
#define ROWS   16      // batch rows per workgroup (== WMMA M)

#define NW_Z   28      // waves computing z tiles (448 / 16)
#define KK_Z   24      // k-steps for z GEMM (96 / 4)
#define NW_I   2       // waves computing intensity tiles (32 / 16)
#define KK_I   16      // k-steps for intensity GEMM (64 / 4)

// LDS partition (in floats)
#define OFF_WS   0                                   // 28*24*32 v2f = 43008 f
#define OFF_WI   (OFF_WS + NW_Z * KK_Z * 32 * 2)     // 2*16*32 v2f  = 2048 f
#define OFF_AS   (OFF_WI + NW_I * KK_I * 32 * 2)     // 16*96        = 1536 f
#define OFF_ZS   (OFF_AS + ROWS * DIN)               // 16*448       = 7168 f
#define OFF_CD   (OFF_ZS + ROWS * NOUT)              // 16*64        = 1024 f
#define OFF_CB   (OFF_CD + ROWS * HIDDEN)            // 16*64        = 1024 f
#define SMEM_FLOATS (OFF_CB + ROWS * HIDDEN)         // 55808 floats = 223232 B

typedef float v2f __attribute__((ext_vector_type(2)));
typedef float v8f __attribute__((ext_vector_type(8)));

__device__ __forceinline__ float sigmoidf_(float x) {
    return 1.0f / (1.0f + __expf(-x));
}
__device__ __forceinline__ float softplusf_(float x) {
    return (x > 20.0f) ? x : log1pf(__expf(x));
}

__global__ __launch_bounds__(1024)
void ctlstm_fwd_kernel(const float* __restrict__ batch,
                       const float* __restrict__ W_rec,
                       const float* __restrict__ b_rec,
                       const float* __restrict__ W_int,
                       const float* __restrict__ b_int,
                       float* __restrict__ out)
{
    extern __shared__ float smem[];
    v2f*   Ws = (v2f*)(smem + OFF_WS);   // swizzled W_rec fragments
    v2f*   Wi = (v2f*)(smem + OFF_WI);   // swizzled W_int fragments
    float* As = smem + OFF_AS;           // [x_t | h], 16x96
    float* Zs = smem + OFF_ZS;           // z, 16x448
    float* CD = smem + OFF_CD;           // c_decay state
    float* CB = smem + OFF_CB;           // c_bar state

    const int tid  = threadIdx.x;
    const int lane = tid & 31;
    const int wave = tid >> 5;
    const int b0   = blockIdx.x * ROWS;

    const int l16 = lane & 15;   // M (A) or N (B,C,D) index
    const int lhi = lane >> 4;   // half-wave select

    // init h, c_decay, c_bar
    for (int e = tid; e < ROWS * HIDDEN; e += blockDim.x) { CD[e] = 0.0f; CB[e] = 0.0f; }
    for (int e = tid; e < ROWS * DIN;    e += blockDim.x) { As[e] = 0.0f; }

    // ---- pre-swizzle W_rec into per-wave WMMA B-fragment layout in LDS ----
    // ISA (32-bit B 4x16): lane l, VGPR v holds B[K = v + 2*(l>>4)][N = l&15]
    float brec_n = 0.0f;
    if (wave < NW_Z) {
        const int ncol = wave * 16 + l16;
        brec_n = b_rec[ncol];
        #pragma unroll
        for (int kk = 0; kk < KK_Z; ++kk) {
            const int krow = kk * 4 + 2 * lhi;
            v2f f;
            f.x = W_rec[(krow + 0) * NOUT + ncol];
            f.y = W_rec[(krow + 1) * NOUT + ncol];
            Ws[(wave * KK_Z + kk) * 32 + lane] = f;
        }
    }
    float bint_n = 0.0f;
    if (wave < NW_I) {
        const int ncol = wave * 16 + l16;
        bint_n = b_int[ncol];
        #pragma unroll
        for (int kk = 0; kk < KK_I; ++kk) {
            const int krow = kk * 4 + 2 * lhi;
            v2f f;
            f.x = W_int[(krow + 0) * NF + ncol];
            f.y = W_int[(krow + 1) * NF + ncol];
            Wi[(wave * KK_I + kk) * 32 + lane] = f;
        }
    }

    // per-wave fragment base pointers (lane-contiguous -> conflict-free b64)
    const v2f* wsw = &Ws[wave * KK_Z * 32 + lane];
    const v2f* wiw = &Wi[wave * KK_I * 32 + lane];
    const float* arow = &As[l16 * DIN];

    // ---- running pointers (replace per-step 64-bit address math) ----
    // staging threads (tid < 512): row = tid>>5, col = tid&31
    const int   srow = tid >> 5;
    const int   scol = tid & 31;
    const float* xp  = batch + ((size_t)(b0 + srow) * SEQ) * 33 + 1 + scol;
    float* axs = &As[srow * DIN + scol];

    // gate-update threads (all 1024): row = tid>>6, j = tid&63
    const int   grow = tid >> 6;
    const int   j    = tid & 63;
    const int   bidx = b0 + grow;
    const float* tp  = batch + ((size_t)bidx * SEQ) * 33;     // times
    float* po  = out + (size_t)0 * BATCH * SEQ * HIDDEN + (size_t)bidx * SEQ * HIDDEN + j;
    float* pc  = out + (size_t)1 * BATCH * SEQ * HIDDEN + (size_t)bidx * SEQ * HIDDEN + j;
    float* pcb = out + (size_t)2 * BATCH * SEQ * HIDDEN + (size_t)bidx * SEQ * HIDDEN + j;
    float* pd  = out + (size_t)3 * BATCH * SEQ * HIDDEN + (size_t)bidx * SEQ * HIDDEN + j;

    // intensity (waves 0..1): lane covers rows m = 8*lhi + r, col = 16*wave + l16
    float* pi = out + (size_t)4 * BATCH * SEQ * HIDDEN
              + (size_t)(b0 + 8 * lhi) * (SEQ - 1) * NF + wave * 16 + l16;

    __syncthreads();

    // prime the x pipeline
    float xreg = 0.0f;
    if (tid < ROWS * NF) xreg = xp[0];

    for (int t = 0; t < SEQ; ++t) {
        // ---- stage x_t; issue load of x_{t+1}; prefetch x_{t+2} ----
        if (tid < ROWS * NF) {
            *axs = xreg;                              // wait lands here, 1 step late
            if (t < SEQ - 1) xreg = xp[33];           // in flight during this step
            if (t < SEQ - 2) __builtin_prefetch(xp + 66, 0, 1);
            xp += 33;
        }
        __syncthreads();   // (A) x_t and h ready

        // ---- z = [x|h] @ W_rec + b_rec via fp32 WMMA, 24 k-steps ----
        if (wave < NW_Z) {
            v8f acc;
            #pragma unroll
            for (int r = 0; r < 8; ++r) acc[r] = brec_n;
            const int ncol = wave * 16 + l16;
            #pragma unroll
            for (int kk = 0; kk < KK_Z; ++kk) {
                const int kbase = kk * 4 + 2 * lhi;
                v2f a;
                a.x = arow[kbase + 0];
                a.y = arow[kbase + 1];
                const v2f b = wsw[kk * 32];
                acc = __builtin_amdgcn_wmma_f32_16x16x4_f32(
                    /*neg_a=*/false, a, /*neg_b=*/false, b,
                    /*c_mod=*/(short)0, acc, /*reuse_a=*/false, /*reuse_b=*/false);
            }
            #pragma unroll
            for (int r = 0; r < 8; ++r) {
                const int m = r + 8 * lhi;           // C/D row layout
                Zs[m * NOUT + ncol] = acc[r];
            }
        }
        __syncthreads();   // (B) z ready; all reads of old h done

        // ---- gate & state update: one thread per (row, hidden j) ----
        {
            const float* zrow = &Zs[grow * NOUT];
            const float i_  = sigmoidf_(zrow[          j]);
            const float f_  = sigmoidf_(zrow[ 64 +     j]);
            const float g_  = tanhf    (zrow[128 +     j]);
            const float o_  = sigmoidf_(zrow[192 +     j]);
            const float ib_ = sigmoidf_(zrow[256 +     j]);
            const float fb_ = sigmoidf_(zrow[320 +     j]);
            const float d_  = softplusf_(zrow[384 +    j]);

            const float dt = (t < SEQ - 1) ? (tp[33] - tp[0]) : 0.0f;
            tp += 33;

            const int e = grow * HIDDEN + j;
            const float cd     = CD[e];
            const float cb     = CB[e];
            const float c      = f_ * cd + i_ * g_;
            const float cbar_n = fb_ * cb + ib_ * g_;
            const float cdec_n = cbar_n + (c - cbar_n) * __expf(-d_ * dt);
            const float hn     = o_ * tanhf(cdec_n);
            CD[e] = cdec_n;
            CB[e] = cbar_n;
            As[grow * DIN + NF + j] = hn;            // new h for next step + intensity

            po[0]  = o_;      po  += HIDDEN;
            pc[0]  = c;       pc  += HIDDEN;
            pcb[0] = cbar_n;  pcb += HIDDEN;
            pd[0]  = d_;      pd  += HIDDEN;
        }
        __syncthreads();   // (C) h_n resident in As

        // ---- intensity = softplus(h_n @ W_int + b_int); drop last step ----
        if (wave < NW_I && t < SEQ - 1) {
            v8f acc;
            #pragma unroll
            for (int r = 0; r < 8; ++r) acc[r] = bint_n;
            #pragma unroll
            for (int kk = 0; kk < KK_I; ++kk) {
                const int kbase = kk * 4 + 2 * lhi;
                v2f a;
                a.x = arow[NF + kbase + 0];
                a.y = arow[NF + kbase + 1];
                const v2f b = wiw[kk * 32];
                acc = __builtin_amdgcn_wmma_f32_16x16x4_f32(
                    false, a, false, b, (short)0, acc, false, false);
            }
            #pragma unroll
            for (int r = 0; r < 8; ++r) {
                pi[(size_t)r * (SEQ - 1) * NF] = softplusf_(acc[r]);
            }
            pi += NF;
        }
        // No trailing barrier needed: next iteration's x staging touches only
        // As[:,0:32] (disjoint from the h region read above), and barrier (A)
        // of the next iteration orders staging vs. the next WMMA.
    }
}

extern "C" void kernel_launch(void* const* d_in, const int* in_sizes, int n_in,
                              void* d_out, int out_size, void* d_ws, size_t ws_size,
                              hipStream_t stream) {
    const float* batch = (const float*)d_in[0];
    const float* W_rec = (const float*)d_in[1];
    const float* b_rec = (const float*)d_in[2];
    const float* W_int = (const float*)d_in[3];
    const float* b_int = (const float*)d_in[4];
    float* out = (float*)d_out;

    ctlstm_fwd_kernel<<<BATCH / ROWS, 1024, SMEM_FLOATS * sizeof(float), stream>>>(
        batch, W_rec, b_rec, W_int, b_int, out);
}